// KrausRK_3444563771811
// MI455X (gfx1250) — compile-verified
//
#include <hip/hip_runtime.h>
#include <hip/hip_bf16.h>

typedef __attribute__((ext_vector_type(2))) float v2f;
typedef __attribute__((ext_vector_type(8))) float v8f;

#define NDIM 64
#define LDSZ 65              // padded LDS row stride (bank-conflict-free columns)
#define MATS (NDIM * LDSZ)   // 4160 floats per LDS matrix
#define DT 0.01f

// One 64x64x64 fp32 matmul done cooperatively by 8 waves (256 threads).
// Tiles: 4 rows x 4 cols of 16x16; wave w owns tile row (w&3) and col pair (w>>2).
// A: MxK, row-major, stride lda (65 for LDS, 64 for global).
// B: if !BT, KxN stride ldb; if BT, B holds M (NxK view) and we compute X @ M^T.
// ACC: accumulate into existing dst contents.
template <bool BT, bool ACC>
__device__ __forceinline__ void mm64(float* __restrict__ dst, int ldd,
                                     const float* __restrict__ A, int lda,
                                     const float* __restrict__ B, int ldb,
                                     int wave, int lane) {
  const int lanelo = lane & 15;
  const int grp    = lane >> 4;      // lane group: K/M offset selector
  const int r  = (wave & 3) << 4;    // tile row base
  const int c0 = (wave >> 2) << 5;   // first tile col base
  const int c1 = c0 + 16;

  v8f acc0, acc1;
  if constexpr (ACC) {
#pragma unroll
    for (int v = 0; v < 8; ++v) {
      const int m = r + v + 8 * grp;
      acc0[v] = dst[m * ldd + c0 + lanelo];
      acc1[v] = dst[m * ldd + c1 + lanelo];
    }
  } else {
#pragma unroll
    for (int v = 0; v < 8; ++v) { acc0[v] = 0.0f; acc1[v] = 0.0f; }
  }

  const int am = r + lanelo;  // A row for this lane (M = lane%16 within tile)
#pragma unroll
  for (int kk = 0; kk < 16; ++kk) {
    const int kb = (kk << 2) + 2 * grp;  // K = kb + v, v in {0,1}
    v2f a, b0, b1;
    a[0] = A[am * lda + kb];
    a[1] = A[am * lda + kb + 1];
    if constexpr (!BT) {
      b0[0] = B[kb * ldb + c0 + lanelo];
      b0[1] = B[(kb + 1) * ldb + c0 + lanelo];
      b1[0] = B[kb * ldb + c1 + lanelo];
      b1[1] = B[(kb + 1) * ldb + c1 + lanelo];
    } else {  // B_frag[k][n] = M[n][k]
      b0[0] = B[(c0 + lanelo) * ldb + kb];
      b0[1] = B[(c0 + lanelo) * ldb + kb + 1];
      b1[0] = B[(c1 + lanelo) * ldb + kb];
      b1[1] = B[(c1 + lanelo) * ldb + kb + 1];
    }
    acc0 = __builtin_amdgcn_wmma_f32_16x16x4_f32(false, a, false, b0,
                                                 (short)0, acc0, false, false);
    acc1 = __builtin_amdgcn_wmma_f32_16x16x4_f32(false, a, false, b1,
                                                 (short)0, acc1, false, false);
  }

#pragma unroll
  for (int v = 0; v < 8; ++v) {
    const int m = r + v + 8 * grp;
    dst[m * ldd + c0 + lanelo] = acc0[v];
    dst[m * ldd + c1 + lanelo] = acc1[v];
  }
}

// Precompute P21 = U_one @ (I + W + W^2), W = I - U_half.  Single workgroup.
__global__ void __launch_bounds__(256) p21_kernel(const float* __restrict__ Uh,
                                                  const float* __restrict__ Uo,
                                                  float* __restrict__ p21) {
  __shared__ float sm[3 * MATS];
  float* W  = sm;
  float* W2 = sm + MATS;
  float* T  = sm + 2 * MATS;
  const int t = threadIdx.x, wave = t >> 5, lane = t & 31;

#pragma unroll
  for (int i = 0; i < 16; ++i) {
    const int e = t + i * 256, rr = e >> 6, cc = e & 63;
    W[rr * LDSZ + cc] = (rr == cc ? 1.0f : 0.0f) - Uh[e];
  }
  __syncthreads();
  mm64<false, false>(W2, LDSZ, W, LDSZ, W, LDSZ, wave, lane);  // W2 = W@W
  __syncthreads();
#pragma unroll
  for (int i = 0; i < 16; ++i) {
    const int e = t + i * 256, rr = e >> 6, cc = e & 63, o = rr * LDSZ + cc;
    W[o] = (rr == cc ? 1.0f : 0.0f) + W[o] + W2[o];            // inv ≈ I+W+W^2
  }
  __syncthreads();
  mm64<false, false>(T, LDSZ, Uo, NDIM, W, LDSZ, wave, lane);  // T = Uo @ inv
  __syncthreads();
#pragma unroll
  for (int i = 0; i < 16; ++i) {
    const int e = t + i * 256;
    p21[e] = T[(e >> 6) * LDSZ + (e & 63)];
  }
}

// Main RK3 Kraus step: one workgroup (8 waves) per density matrix.
__global__ void __launch_bounds__(256) kraus_kernel(
    const float* __restrict__ rho0, const float* __restrict__ Uh,
    const float* __restrict__ Uo, const float* __restrict__ L0,
    const float* __restrict__ Lh, const float* __restrict__ L1,
    const float* __restrict__ P21, float* __restrict__ out) {
  __shared__ float sm[6 * MATS];  // ~97.5 KB: 3 workgroups / 320KB WGP
  float* R  = sm;            // rho  -> later reused as D1
  float* D0 = sm + MATS;     // D0   -> later reused as Z0
  float* T1 = sm + 2 * MATS; // matmul temp
  float* T2 = sm + 3 * MATS; // X    -> later stage2
  float* B4 = sm + 4 * MATS; // Rh (stage1) -> later S
  float* B5 = sm + 5 * MATS; // Dh   -> later Y

  const int b = blockIdx.x, t = threadIdx.x;
  const int wave = t >> 5, lane = t & 31;
  const float* rb = rho0 + (size_t)b * (NDIM * NDIM);

#pragma unroll
  for (int i = 0; i < 16; ++i) {
    const int e = t + i * 256;
    R[(e >> 6) * LDSZ + (e & 63)] = rb[e];
  }
  __syncthreads();

  // ---- D0 = sum_k L0k rho L0k^T ----
  mm64<false, false>(T1, LDSZ, L0,        NDIM, R,  LDSZ, wave, lane); __syncthreads();
  mm64<true,  false>(D0, LDSZ, T1,        LDSZ, L0, NDIM, wave, lane); __syncthreads();
  mm64<false, false>(T1, LDSZ, L0 + 4096, NDIM, R,  LDSZ, wave, lane); __syncthreads();
  mm64<true,  true >(D0, LDSZ, T1, LDSZ, L0 + 4096, NDIM, wave, lane); __syncthreads();

  // ---- X = rho + 0.5*DT*D0 ; stage1 = Uh X Uh^T ----
#pragma unroll
  for (int i = 0; i < 16; ++i) {
    const int e = t + i * 256, o = (e >> 6) * LDSZ + (e & 63);
    T2[o] = R[o] + (0.5f * DT) * D0[o];
  }
  __syncthreads();
  mm64<false, false>(T1, LDSZ, Uh, NDIM, T2, LDSZ, wave, lane); __syncthreads();
  mm64<true,  false>(B4, LDSZ, T1, LDSZ, Uh, NDIM, wave, lane); __syncthreads();

  // ---- Dh = sum_k Lhk stage1 Lhk^T ----
  mm64<false, false>(T1, LDSZ, Lh,        NDIM, B4, LDSZ, wave, lane); __syncthreads();
  mm64<true,  false>(B5, LDSZ, T1,        LDSZ, Lh, NDIM, wave, lane); __syncthreads();
  mm64<false, false>(T1, LDSZ, Lh + 4096, NDIM, B4, LDSZ, wave, lane); __syncthreads();
  mm64<true,  true >(B5, LDSZ, T1, LDSZ, Lh + 4096, NDIM, wave, lane); __syncthreads();

  // ---- Y = P21 Dh P21^T (reuses B5) ----
  mm64<false, false>(T1, LDSZ, P21, NDIM, B5,  LDSZ, wave, lane); __syncthreads();
  mm64<true,  false>(B5, LDSZ, T1,  LDSZ, P21, NDIM, wave, lane); __syncthreads();

  // ---- S = Uo rho Uo^T (into B4, Rh is dead) ----
  mm64<false, false>(T1, LDSZ, Uo, NDIM, R,  LDSZ, wave, lane); __syncthreads();
  mm64<true,  false>(B4, LDSZ, T1, LDSZ, Uo, NDIM, wave, lane); __syncthreads();

  // ---- stage2 = S + DT*Y ----
#pragma unroll
  for (int i = 0; i < 16; ++i) {
    const int e = t + i * 256, o = (e >> 6) * LDSZ + (e & 63);
    T2[o] = B4[o] + DT * B5[o];
  }
  __syncthreads();

  // ---- D1 = sum_k L1k stage2 L1k^T (into R, rho is dead) ----
  mm64<false, false>(T1, LDSZ, L1,        NDIM, T2, LDSZ, wave, lane); __syncthreads();
  mm64<true,  false>(R,  LDSZ, T1,        LDSZ, L1, NDIM, wave, lane); __syncthreads();
  mm64<false, false>(T1, LDSZ, L1 + 4096, NDIM, T2, LDSZ, wave, lane); __syncthreads();
  mm64<true,  true >(R,  LDSZ, T1, LDSZ, L1 + 4096, NDIM, wave, lane); __syncthreads();

  // ---- Z0 = Uo D0 Uo^T (into D0) ----
  mm64<false, false>(T1, LDSZ, Uo, NDIM, D0, LDSZ, wave, lane); __syncthreads();
  mm64<true,  false>(D0, LDSZ, T1, LDSZ, Uo, NDIM, wave, lane); __syncthreads();

  // ---- out = S + DT*(Z0/6 + 2Y/3 + D1/6) ----
  float* ob = out + (size_t)b * (NDIM * NDIM);
#pragma unroll
  for (int i = 0; i < 16; ++i) {
    const int e = t + i * 256, o = (e >> 6) * LDSZ + (e & 63);
    ob[e] = B4[o] + DT * ((1.0f / 6.0f) * D0[o] + (2.0f / 3.0f) * B5[o] +
                          (1.0f / 6.0f) * R[o]);
  }
}

extern "C" void kernel_launch(void* const* d_in, const int* in_sizes, int n_in,
                              void* d_out, int out_size, void* d_ws, size_t ws_size,
                              hipStream_t stream) {
  const float* rho0 = (const float*)d_in[0];
  const float* Uh   = (const float*)d_in[1];
  const float* Uo   = (const float*)d_in[2];
  const float* L0   = (const float*)d_in[3];
  const float* Lh   = (const float*)d_in[4];
  const float* L1   = (const float*)d_in[5];
  float* p21 = (float*)d_ws;  // 64*64 floats of scratch

  p21_kernel<<<1, 256, 0, stream>>>(Uh, Uo, p21);
  kraus_kernel<<<4096, 256, 0, stream>>>(rho0, Uh, Uo, L0, Lh, L1, p21,
                                         (float*)d_out);
}